// compress_jpeg_24661702214024
// MI455X (gfx1250) — compile-verified
//
#include <hip/hip_runtime.h>
#include <hip/hip_bf16.h>
#include <hip/hip_fp16.h>

typedef __attribute__((ext_vector_type(2)))  float    v2f;
typedef __attribute__((ext_vector_type(8)))  float    v8f;

// Probe WMMA builtins only in the DEVICE pass; the host pass has no
// __builtin_amdgcn_* and must not see them at all.
#if defined(__HIP_DEVICE_COMPILE__)
#  if __has_builtin(__builtin_amdgcn_wmma_f32_16x16x4_f32)
#    define USE_F32_WMMA 1
#  else
#    error "f32 WMMA builtin missing on this toolchain (device pass)"
#  endif
#endif

#define PI_16 0.19634954084936207f  // pi/16

// Stage 1: acc += Cconst(16x16) * B, B supplied by getB(k,n) (LDS reads).
// Result in WMMA C/D layout: acc[r] = D[r + 8*(lane>>4)][lane&15].
template <typename FB>
__device__ __forceinline__ v8f mac16_constA(const v2f (&cf)[4], int lane,
                                            FB getB, v8f acc) {
#if defined(__HIP_DEVICE_COMPILE__)
    const int half = lane >> 4;
    const int n    = lane & 15;
#pragma unroll
    for (int i = 0; i < 4; ++i) {
        const int ka = 4 * i + 2 * half;
        v2f b;
        b.x = getB(ka, n);
        b.y = getB(ka + 1, n);
        acc = __builtin_amdgcn_wmma_f32_16x16x4_f32(
            false, cf[i], false, b, (short)0, acc, false, false);
    }
#endif
    return acc;
}

// Stage 2: acc += A * Cconst(16x16), A supplied by getA(m,k) (LDS reads).
template <typename FA>
__device__ __forceinline__ v8f mac16_constB(FA getA, const v2f (&cf)[4],
                                            int lane, v8f acc) {
#if defined(__HIP_DEVICE_COMPILE__)
    const int half = lane >> 4;
    const int m    = lane & 15;
#pragma unroll
    for (int i = 0; i < 4; ++i) {
        const int ka = 4 * i + 2 * half;
        v2f a;
        a.x = getA(m, ka);
        a.y = getA(m, ka + 1);
        acc = __builtin_amdgcn_wmma_f32_16x16x4_f32(
            false, a, false, cf[i], (short)0, acc, false, false);
    }
#endif
    return acc;
}

// One wave (32 lanes) per 32x32 pixel tile.
__global__ __launch_bounds__(32) void jpeg_compress_kernel(
    const float* __restrict__ img,      // [32,3,512,512]
    const float* __restrict__ mat,      // [3,3] (rgb -> ycbcr, row=c, col=d)
    const float* __restrict__ shift,    // [3]
    const float* __restrict__ scaleT,   // [8,8]
    const float* __restrict__ yT,       // [8,8]
    const float* __restrict__ cT,       // [8,8]
    const int*   __restrict__ factorP,  // scalar
    float* __restrict__ out)            // y | cb | cr concatenated
{
    constexpr int H = 512, W = 512;
    constexpr int PF = 40;   // full-res LDS pitch: rows k,k+2 differ by
                             // 80 mod 64 = 16 banks -> conflict-free halves;
                             // 160B rows keep float4 stores 16B-aligned.
    constexpr int PH = 17;   // 16-wide LDS pitch (rows 34 mod 64 apart; and
                             // 17 coprime 64 -> conflict-free column reads)
    constexpr size_t Y_BASE  = 0;
    constexpr size_t CB_BASE = (size_t)32 * 4096 * 64;            // 8388608
    constexpr size_t CR_BASE = CB_BASE + (size_t)32 * 1024 * 64;  // +2097152

    __shared__ float sY[32 * PF];    // Y - 128, full res
    __shared__ float sCbF[32 * PF];  // Cb - 128, full res
    __shared__ float sCrF[32 * PF];  // Cr - 128, full res
    __shared__ float sCb[16 * PH];   // pooled
    __shared__ float sCr[16 * PH];
    __shared__ float sT[16 * PH];    // stage-1 DCT temp

    const int lane  = threadIdx.x;           // 0..31
    const int tileX = blockIdx.x;            // 0..15
    const int tileY = blockIdx.y;            // 0..15
    const int b     = blockIdx.z;            // 0..31
    const int px0   = tileX * 32;
    const int py0   = tileY * 32;

    // ---- Phase A: 128-bit loads, color convert, 128-bit LDS stores ----
    const float m00 = mat[0], m01 = mat[1], m02 = mat[2];
    const float m10 = mat[3], m11 = mat[4], m12 = mat[5];
    const float m20 = mat[6], m21 = mat[7], m22 = mat[8];
    const float s0 = shift[0] - 128.0f;
    const float s1 = shift[1] - 128.0f;
    const float s2 = shift[2] - 128.0f;

    const size_t imgB = (size_t)b * 3 * H * W;
    const int rsub = lane >> 3;        // 4 rows covered per iteration
    const int csub = (lane & 7) * 4;   // each lane owns 4 consecutive pixels
    for (int i = 0; i < 8; ++i) {
        const int row  = i * 4 + rsub;
        const size_t p = imgB + (size_t)(py0 + row) * W + (px0 + csub);
        const float4 R  = *(const float4*)(img + p);
        const float4 G  = *(const float4*)(img + p + (size_t)H * W);
        const float4 Bv = *(const float4*)(img + p + (size_t)2 * H * W);
        float4 yv, cbv, crv;
#pragma unroll
        for (int j = 0; j < 4; ++j) {
            const float r  = (&R.x)[j]  * 255.0f;
            const float g  = (&G.x)[j]  * 255.0f;
            const float bl = (&Bv.x)[j] * 255.0f;
            (&yv.x)[j]  = fmaf(r, m00, fmaf(g, m10, fmaf(bl, m20, s0)));
            (&cbv.x)[j] = fmaf(r, m01, fmaf(g, m11, fmaf(bl, m21, s1)));
            (&crv.x)[j] = fmaf(r, m02, fmaf(g, m12, fmaf(bl, m22, s2)));
        }
        *(float4*)(sY   + row * PF + csub) = yv;
        *(float4*)(sCbF + row * PF + csub) = cbv;
        *(float4*)(sCrF + row * PF + csub) = crv;
    }
    __syncthreads();

    // ---- 2x2 average pool for chroma (256 values each, 8 per lane) ----
    for (int t = 0; t < 8; ++t) {
        const int idx = t * 32 + lane;
        const int i2 = idx >> 4, j2 = idx & 15;
        const int o = (2 * i2) * PF + 2 * j2;
        sCb[i2 * PH + j2] = 0.25f * (sCbF[o] + sCbF[o + 1] + sCbF[o + PF] + sCbF[o + PF + 1]);
        sCr[i2 * PH + j2] = 0.25f * (sCrF[o] + sCrF[o + 1] + sCrF[o + PF] + sCrF[o + PF + 1]);
    }
    __syncthreads();

    // ---- per-lane quantization factors: scale[u,v] / (table[u,v]*f) ----
    const float fq = (float)factorP[0];
    const int v = lane & 7;                 // output column within 8x8 block
    v8f fcY, fcC;
#pragma unroll
    for (int r = 0; r < 8; ++r) {
        const int idx = r * 8 + v;
        fcY[r] = scaleT[idx] / (yT[idx] * fq);
        fcC[r] = scaleT[idx] / (cT[idx] * fq);
    }

    const int half = lane >> 4;             // D-layout row half (qy)
    const int m    = lane & 15;
    const int qx   = (lane >> 3) & 1;       // output block column parity

    // ---- Per-lane DCT-basis fragment, computed ONCE, branch-free. ----
    // Stage-1 A-frag == stage-2 B-frag per lane:
    //   Ct16[m][ka] = Cb16[ka][m] = mask * cos((2*(ka&7)+1)*(m&7)*pi/16)
    // cos computed unconditionally -> v_cndmask select, no EXEC branching.
    v2f cf[4];
#pragma unroll
    for (int i = 0; i < 4; ++i) {
        const int ka = 4 * i + 2 * half;    // even, so ka and ka+1 share octant
        const float c0 = __cosf((float)((2 * (ka & 7) + 1) * (m & 7)) * PI_16);
        const float c1 = __cosf((float)((2 * ((ka + 1) & 7) + 1) * (m & 7)) * PI_16);
        const bool same = ((m >> 3) == (ka >> 3));
        cf[i].x = same ? c0 : 0.0f;
        cf[i].y = same ? c1 : 0.0f;
    }

    // ---- Phase B: 6 tile-DCT units (4 Y + Cb + Cr), WMMA both stages ----
#pragma unroll
    for (int unit = 0; unit < 6; ++unit) {
        const float* plane;
        int pitch, r0, c0;
        size_t blk, base;
        if (unit < 4) {
            const int uy = unit >> 1, ux = unit & 1;
            plane = sY; pitch = PF; r0 = uy * 16; c0 = ux * 16;
            base = Y_BASE;
            blk = (size_t)b * 4096
                + (size_t)(tileY * 4 + uy * 2 + half) * 64
                + (size_t)(tileX * 4 + ux * 2 + qx);
        } else {
            plane = (unit == 4) ? sCb : sCr;
            pitch = PH; r0 = 0; c0 = 0;
            base = (unit == 4) ? CB_BASE : CR_BASE;
            blk = (size_t)b * 1024
                + (size_t)(tileY * 2 + half) * 32
                + (size_t)(tileX * 2 + qx);
        }
        const v8f fcv = (unit < 4) ? fcY : fcC;   // register select, no spill

        // Stage 1: T = blockdiag(C^T) * A   (pixel rows as B operand)
        v8f t = {};
        t = mac16_constA(cf, lane,
                         [&](int k, int n) -> float {
                             return plane[(r0 + k) * pitch + (c0 + n)];
                         },
                         t);

        // D-layout -> plain 16x16 in LDS (pitch 17 => conflict-free col reads)
        __syncthreads();
#pragma unroll
        for (int r = 0; r < 8; ++r)
            sT[(r + 8 * half) * PH + m] = t[r];
        __syncthreads();

        // Stage 2: D = T * blockdiag(C)     (T columns as A operand)
        v8f d = {};
        d = mac16_constB([&](int mm, int k) -> float { return sT[mm * PH + k]; },
                         cf, lane, d);

        // Quantize + round (v_rndne = RNE matches jnp.round) and store.
        // lane -> (qy=half, qx, v); vgpr r -> u.
        float* o = out + base + blk * 64 + v;
#pragma unroll
        for (int r = 0; r < 8; ++r)
            o[r * 8] = rintf(d[r] * fcv[r]);
        __syncthreads();  // protect sT before next unit
    }
}

extern "C" void kernel_launch(void* const* d_in, const int* in_sizes, int n_in,
                              void* d_out, int out_size, void* d_ws, size_t ws_size,
                              hipStream_t stream) {
    const float* img    = (const float*)d_in[0];
    const float* mat    = (const float*)d_in[1];
    const float* shift  = (const float*)d_in[2];
    // d_in[3] = dct_tensor (unused: basis computed analytically)
    const float* scaleT = (const float*)d_in[4];
    const float* yT     = (const float*)d_in[5];
    const float* cT     = (const float*)d_in[6];
    const int*   factor = (const int*)d_in[7];
    float* out = (float*)d_out;

    dim3 grid(16, 16, 32);   // W/32, H/32, B
    dim3 block(32);          // one wave per 32x32 tile (EXEC all-ones for WMMA)
    jpeg_compress_kernel<<<grid, block, 0, stream>>>(
        img, mat, shift, scaleT, yT, cT, factor, out);
}